// trisum_57423712747823
// MI455X (gfx1250) — compile-verified
//
#include <hip/hip_runtime.h>

// ---------------------------------------------------------------------------
// out = sum_m weights[idx[m,0]] * input[idx[m,1]] * input[idx[m,2]]
// M = 16,777,216 triples; all indices < 2048.
// HBM-stream bound on the 192 MiB idxs array (~8.6 us at 23.3 TB/s).
// Tables (8 KB each) staged to LDS via CDNA5 async global->LDS copies,
// gathers served from LDS, block reduce finished with v_wmma_f32_16x16x4_f32.
// ---------------------------------------------------------------------------

#define STATE      2048
#define NBLOCKS    2048
#define NTHREADS   256
#define ITERS      8        // 8 iters * 4 triples = 32 triples/thread
// NBLOCKS * NTHREADS * 32 = 16,777,216 triples exactly.

typedef __attribute__((ext_vector_type(2))) float v2f;
typedef __attribute__((ext_vector_type(8))) float v8f;

// Native clang vector type the async-copy builtin expects (NOT HIP's int4
// class), qualified with the global (AS1) / LDS (AS3) address spaces.
typedef int nvi4 __attribute__((vector_size(16)));
typedef __attribute__((address_space(1))) nvi4 gvi4;
typedef __attribute__((address_space(3))) nvi4 lvi4;

__global__ __launch_bounds__(NTHREADS) void trisum_main_kernel(
    const float* __restrict__ input,
    const float* __restrict__ weights,
    const int*   __restrict__ idxs,
    float*       __restrict__ partials) {
  __shared__ float sW[STATE];
  __shared__ float sI[STATE];
  __shared__ float sred[NTHREADS];

  const int tid = threadIdx.x;

  // ---- Stage gather tables (8 KB each) into LDS --------------------------
#if __has_builtin(__builtin_amdgcn_global_load_async_to_lds_b128)
  {
    char* wB = (char*)weights;   // builtin takes non-const pointers
    char* iB = (char*)input;
#pragma unroll
    for (int j = 0; j < (STATE * 4 / 16) / NTHREADS; ++j) {   // 2 chunks/thread
      const int v = tid + j * NTHREADS;                       // 16-byte chunk id
      __builtin_amdgcn_global_load_async_to_lds_b128(
          (gvi4*)(wB + v * 16), (lvi4*)&sW[v * 4], 0, 0);
      __builtin_amdgcn_global_load_async_to_lds_b128(
          (gvi4*)(iB + v * 16), (lvi4*)&sI[v * 4], 0, 0);
    }
    asm volatile("s_wait_asynccnt 0" ::: "memory");
  }
  __syncthreads();
#else
  for (int j = tid; j < STATE; j += NTHREADS) {
    sW[j] = weights[j];
    sI[j] = input[j];
  }
  __syncthreads();
#endif

  // ---- Stream idxs with dense b128 loads, gather from LDS -----------------
  const int blockStart = blockIdx.x * (NTHREADS * ITERS * 4);  // first triple
  float acc = 0.0f;
#pragma unroll
  for (int i = 0; i < ITERS; ++i) {
    const int tripleBase = blockStart + ((i * NTHREADS + tid) << 2); // 4 triples
    const int4* p = (const int4*)(idxs + tripleBase * 3);  // 48B, 16B-aligned
    const int4 q0 = p[0];
    const int4 q1 = p[1];
    const int4 q2 = p[2];
    acc += sW[q0.x] * sI[q0.y] * sI[q0.z];
    acc += sW[q0.w] * sI[q1.x] * sI[q1.y];
    acc += sW[q1.z] * sI[q1.w] * sI[q2.x];
    acc += sW[q2.y] * sI[q2.z] * sI[q2.w];
  }

  // ---- Block reduction: 256 -> 32 (LDS) -> 16 (WMMA) -> 1 (shuffles) ------
  sred[tid] = acc;
  __syncthreads();
  if (tid < 32) {  // wave 0 only: EXEC all ones within the wave (wave32)
    float x = 0.0f;
#pragma unroll
    for (int k = 0; k < NTHREADS / 32; ++k) x += sred[tid + 32 * k];

#if __has_builtin(__builtin_amdgcn_wmma_f32_16x16x4_f32)
    // A = 16x4 ones, B holds the 32 lane partials in one B VGPR (other = 0).
    // D[m][n] = sum_k B[k][n] = x[n] + x[n+16] for every row m, exact in f32.
    v2f A; A[0] = 1.0f; A[1] = 1.0f;
    v2f B; B[0] = x;    B[1] = 0.0f;
    v8f C = {};
    C = __builtin_amdgcn_wmma_f32_16x16x4_f32(
        /*neg_a=*/false, A, /*neg_b=*/false, B,
        /*c_mod=*/(short)0, C, /*reuse_a=*/false, /*reuse_b=*/false);
    float y = C[0];  // lanes 0..15 hold the 16 pairwise sums
#pragma unroll
    for (int off = 8; off >= 1; off >>= 1) y += __shfl_xor(y, off, 32);
#else
    float y = x;
#pragma unroll
    for (int off = 16; off >= 1; off >>= 1) y += __shfl_xor(y, off, 32);
#endif
    if (tid == 0) partials[blockIdx.x] = y;
  }
}

__global__ __launch_bounds__(256) void trisum_reduce_kernel(
    const float* __restrict__ partials, float* __restrict__ out, int n) {
  __shared__ float s[256];
  float acc = 0.0f;
  for (int i = threadIdx.x; i < n; i += 256) acc += partials[i];
  s[threadIdx.x] = acc;
  __syncthreads();
#pragma unroll
  for (int stride = 128; stride > 0; stride >>= 1) {
    if (threadIdx.x < stride) s[threadIdx.x] += s[threadIdx.x + stride];
    __syncthreads();
  }
  if (threadIdx.x == 0) out[0] = s[0];
}

extern "C" void kernel_launch(void* const* d_in, const int* in_sizes, int n_in,
                              void* d_out, int out_size, void* d_ws, size_t ws_size,
                              hipStream_t stream) {
  (void)in_sizes; (void)n_in; (void)out_size; (void)ws_size;
  const float* input   = (const float*)d_in[0];  // [2048] f32
  const float* weights = (const float*)d_in[1];  // [3*2048^2] f32 (only [0,2048) used)
  const int*   idxs    = (const int*)d_in[2];    // [M,3] i32
  // d_in[3] (idxs_bw) is dead for the forward pass.
  float* out      = (float*)d_out;
  float* partials = (float*)d_ws;                // 2048 floats = 8 KB scratch

  trisum_main_kernel<<<NBLOCKS, NTHREADS, 0, stream>>>(input, weights, idxs, partials);
  trisum_reduce_kernel<<<1, 256, 0, stream>>>(partials, out, NBLOCKS);
}